// Mamba_68882685493650
// MI455X (gfx1250) — compile-verified
//
#include <hip/hip_runtime.h>
#include <cstddef>

#define NUM_BLOCKS 2
#define IN_DIM   128
#define OUT_DIM  10
#define D_MODEL  256
#define D_STATE  16
#define D_CONV   4
#define D_INNER  512
#define DT_RANK  16
#define BATCH    4
#define SEQ      2048
#define BL       (BATCH * SEQ)        // 8192 rows

typedef float v2f __attribute__((ext_vector_type(2)));
typedef float v8f __attribute__((ext_vector_type(8)));

// ---------------------------------------------------------------------------
// Epilogue modes for the WMMA GEMM
// ---------------------------------------------------------------------------
#define EPI_NONE       0
#define EPI_BIAS       1
#define EPI_SOFTPLUS   2   // softplus(acc + bias)
#define EPI_GELU_SKIP  3   // gelu(acc) + skip[row,col]

__device__ __forceinline__ float softplus_f(float x) {
    return (x > 20.0f) ? x : log1pf(expf(x));
}
__device__ __forceinline__ float gelu_tanh_f(float x) {
    // jax.nn.gelu default (approximate=True)
    float x3 = x * x * x;
    return 0.5f * x * (1.0f + tanhf(0.7978845608028654f * (x + 0.044715f * x3)));
}
__device__ __forceinline__ float silu_f(float x) {
    return x / (1.0f + expf(-x));
}

// ---------------------------------------------------------------------------
// WMMA fp32 GEMM:  Out[M,N] = A[M,K] * W[N,K]^T  (+ fused epilogue)
// One wave computes a (16*MSUB) x (16*NSUB) tile with V_WMMA_F32_16X16X4_F32.
// Two-stage software pipeline over K (K must be a multiple of 8) so fragment
// loads for stage k+4 are in flight while the WMMAs consume stage k.
// A-fragment (32-bit 16x4): lanes 0-15 -> K = {k,k+1}; lanes 16-31 -> {k+2,k+3}.
// B-fragment mirrors it; C/D: VGPR i = (M=i, lanes0-15) / (M=i+8, lanes16-31).
// ---------------------------------------------------------------------------
template <int EPI, int MSUB, int NSUB>
__global__ void gemm_wmma_f32(const float* __restrict__ A, int lda,
                              const float* __restrict__ W, int ldw,
                              const float* __restrict__ bias,
                              const float* __restrict__ skip, int ldskip,
                              float* __restrict__ Out, int ldo,
                              int M, int N, int K)
{
    const int lane = threadIdx.x & 31;
    const int wid  = (blockIdx.x * blockDim.x + threadIdx.x) >> 5;

    const int tilesM  = M / (16 * MSUB);
    const int tilesNg = N / (16 * NSUB);
    if (wid >= tilesM * tilesNg) return;          // wave-uniform: EXEC stays all-1

    const int tm = wid % tilesM;
    const int tn = wid / tilesM;
    const int mBase = tm * 16 * MSUB;
    const int nBase = tn * 16 * NSUB;

    const int r  = lane & 15;     // row (A) / col (W) within a 16-tile
    const int hi = lane >> 4;     // 0: K pair {0,1}; 1: K pair {2,3}

    const float* __restrict__ Arow[MSUB];
#pragma unroll
    for (int mi = 0; mi < MSUB; ++mi)
        Arow[mi] = A + (size_t)(mBase + mi * 16 + r) * lda + (hi << 1);
    const float* __restrict__ Wcol[NSUB];
#pragma unroll
    for (int nj = 0; nj < NSUB; ++nj)
        Wcol[nj] = W + (size_t)(nBase + nj * 16 + r) * ldw + (hi << 1);

    v8f acc[MSUB][NSUB] = {};

    v2f a0[MSUB], b0[NSUB], a1[MSUB], b1[NSUB];

    auto loadStage = [&](v2f* aS, v2f* bS, int kk) {
#pragma unroll
        for (int mi = 0; mi < MSUB; ++mi) {
            aS[mi].x = Arow[mi][kk];
            aS[mi].y = Arow[mi][kk + 1];
        }
#pragma unroll
        for (int nj = 0; nj < NSUB; ++nj) {
            bS[nj].x = Wcol[nj][kk];
            bS[nj].y = Wcol[nj][kk + 1];
        }
    };
    auto mmaStage = [&](const v2f* aS, const v2f* bS) {
#pragma unroll
        for (int mi = 0; mi < MSUB; ++mi)
#pragma unroll
            for (int nj = 0; nj < NSUB; ++nj)
                acc[mi][nj] = __builtin_amdgcn_wmma_f32_16x16x4_f32(
                    false, aS[mi], false, bS[nj], (short)0, acc[mi][nj], false, false);
    };

    // K is a multiple of 8 for every call site.
    loadStage(a0, b0, 0);
    loadStage(a1, b1, 4);
    for (int k = 0; k < K; k += 8) {
        __builtin_prefetch(&Arow[0][k + 64], 0, 3);
        mmaStage(a0, b0);                       // consumes data at k
        if (k + 8 < K)  loadStage(a0, b0, k + 8);
        mmaStage(a1, b1);                       // consumes data at k+4
        if (k + 12 < K) loadStage(a1, b1, k + 12);
    }

#pragma unroll
    for (int mi = 0; mi < MSUB; ++mi) {
#pragma unroll
        for (int nj = 0; nj < NSUB; ++nj) {
            const int col = nBase + nj * 16 + r;
            float bval = 0.0f;
            if (EPI == EPI_BIAS || EPI == EPI_SOFTPLUS) bval = bias[col];
#pragma unroll
            for (int i = 0; i < 8; ++i) {
                const int row = mBase + mi * 16 + i + (hi << 3);
                float v = acc[mi][nj][i] + bval;
                if (EPI == EPI_SOFTPLUS)  v = softplus_f(v);
                if (EPI == EPI_GELU_SKIP) v = gelu_tanh_f(v) + skip[(size_t)row * ldskip + col];
                Out[(size_t)row * ldo + col] = v;
            }
        }
    }
}

// ---------------------------------------------------------------------------
// LayerNorm over last dim (256), one wave per row
// ---------------------------------------------------------------------------
__global__ void ln_kernel(const float* __restrict__ in,
                          const float* __restrict__ w,
                          const float* __restrict__ b,
                          float* __restrict__ out, int rows)
{
    const int wid  = (blockIdx.x * blockDim.x + threadIdx.x) >> 5;
    const int lane = threadIdx.x & 31;
    if (wid >= rows) return;

    const float* r = in + (size_t)wid * D_MODEL;
    float s = 0.0f, s2 = 0.0f;
#pragma unroll
    for (int i = lane; i < D_MODEL; i += 32) {
        float v = r[i];
        s += v; s2 += v * v;
    }
#pragma unroll
    for (int off = 16; off > 0; off >>= 1) {
        s  += __shfl_xor(s,  off, 32);
        s2 += __shfl_xor(s2, off, 32);
    }
    const float mean = s * (1.0f / D_MODEL);
    const float var  = s2 * (1.0f / D_MODEL) - mean * mean;
    const float inv  = rsqrtf(var + 1e-5f);

    float* o = out + (size_t)wid * D_MODEL;
#pragma unroll
    for (int i = lane; i < D_MODEL; i += 32)
        o[i] = (r[i] - mean) * inv * w[i] + b[i];
}

// ---------------------------------------------------------------------------
// Depthwise causal conv (width 4) over seq + SiLU.
// x-half of xz[B,L,2*Di] -> xi[B,L,Di]
// ---------------------------------------------------------------------------
__global__ void conv_silu_kernel(const float* __restrict__ xz,
                                 const float* __restrict__ cw,
                                 const float* __restrict__ cb,
                                 float* __restrict__ xi)
{
    const size_t idx = (size_t)blockIdx.x * blockDim.x + threadIdx.x;
    if (idx >= (size_t)BL * D_INNER) return;
    const int d  = (int)(idx % D_INNER);
    const size_t bl = idx / D_INNER;
    const int l  = (int)(bl % SEQ);
    const size_t brow = bl - l;           // b*SEQ

    float acc = cb[d];
#pragma unroll
    for (int j = 0; j < D_CONV; ++j) {
        const int ls = l - (D_CONV - 1) + j;
        if (ls >= 0)
            acc += cw[d * D_CONV + j] * xz[(brow + ls) * (2 * D_INNER) + d];
    }
    xi[idx] = silu_f(acc);
}

// ---------------------------------------------------------------------------
// Selective scan. One wave handles 2 channels (d, d+1); lane = half*16 + n.
// h[n] recurrence over L, h*C reduced across the 16 state lanes via shfl_xor.
// dy holds delta on input and is overwritten with y*silu(z) in place.
// 2-wide unroll over l so the l+1 loads overlap the serial recurrence chain.
// ---------------------------------------------------------------------------
__global__ void scan_kernel(float* __restrict__ dy,
                            const float* __restrict__ xi,
                            const float* __restrict__ dbl,
                            const float* __restrict__ xz,
                            const float* __restrict__ A_log,
                            const float* __restrict__ Dp)
{
    const int wid  = (blockIdx.x * blockDim.x + threadIdx.x) >> 5;
    const int lane = threadIdx.x & 31;
    if (wid >= (BATCH * D_INNER) / 2) return;

    const int half = lane >> 4;
    const int n    = lane & 15;
    const int gch  = wid * 2 + half;          // global channel: b*D_INNER + d
    const int b    = gch / D_INNER;
    const int d    = gch % D_INNER;

    const float Acoef = -expf(A_log[d * D_STATE + n]);
    const float dcoef = Dp[d];
    const size_t rowBase = (size_t)b * SEQ;
    const int DBLW = DT_RANK + 2 * D_STATE;

    float h = 0.0f;
    for (int l = 0; l < SEQ; l += 2) {
        const size_t bl0 = rowBase + l;
        const size_t bl1 = bl0 + 1;
        // issue all loads for both steps up front
        const float dl0 = dy[bl0 * D_INNER + d];
        const float u0  = xi[bl0 * D_INNER + d];
        const float Bv0 = dbl[bl0 * DBLW + DT_RANK + n];
        const float Cv0 = dbl[bl0 * DBLW + DT_RANK + D_STATE + n];
        const float z0  = xz[bl0 * (2 * D_INNER) + D_INNER + d];
        const float dl1 = dy[bl1 * D_INNER + d];
        const float u1  = xi[bl1 * D_INNER + d];
        const float Bv1 = dbl[bl1 * DBLW + DT_RANK + n];
        const float Cv1 = dbl[bl1 * DBLW + DT_RANK + D_STATE + n];
        const float z1  = xz[bl1 * (2 * D_INNER) + D_INNER + d];

        h = expf(dl0 * Acoef) * h + (dl0 * u0) * Bv0;
        float p = h * Cv0;
        p += __shfl_xor(p, 1, 32);
        p += __shfl_xor(p, 2, 32);
        p += __shfl_xor(p, 4, 32);
        p += __shfl_xor(p, 8, 32);
        float y0 = (p + u0 * dcoef) * silu_f(z0);

        h = expf(dl1 * Acoef) * h + (dl1 * u1) * Bv1;
        float q = h * Cv1;
        q += __shfl_xor(q, 1, 32);
        q += __shfl_xor(q, 2, 32);
        q += __shfl_xor(q, 4, 32);
        q += __shfl_xor(q, 8, 32);
        float y1 = (q + u1 * dcoef) * silu_f(z1);

        if (n == 0) {
            dy[bl0 * D_INNER + d] = y0;
            dy[bl1 * D_INNER + d] = y1;
        }
    }
}

// ---------------------------------------------------------------------------
// Mean pool over seq + decoder + softmax. One block per batch row.
// ---------------------------------------------------------------------------
__global__ void head_kernel(const float* __restrict__ h,
                            const float* __restrict__ dec_w,
                            const float* __restrict__ dec_b,
                            float* __restrict__ out)
{
    __shared__ float pooled[D_MODEL];
    __shared__ float logits[OUT_DIM];
    const int b = blockIdx.x;
    const int c = threadIdx.x;

    float s = 0.0f;
    for (int l = 0; l < SEQ; ++l)
        s += h[((size_t)b * SEQ + l) * D_MODEL + c];
    pooled[c] = s * (1.0f / SEQ);
    __syncthreads();

    if (c < OUT_DIM) {
        float acc = dec_b[c];
        for (int k = 0; k < D_MODEL; ++k)
            acc += pooled[k] * dec_w[c * D_MODEL + k];
        logits[c] = acc;
    }
    __syncthreads();

    if (c == 0) {
        float mx = logits[0];
        for (int o = 1; o < OUT_DIM; ++o) mx = fmaxf(mx, logits[o]);
        float e[OUT_DIM], sum = 0.0f;
        for (int o = 0; o < OUT_DIM; ++o) { e[o] = expf(logits[o] - mx); sum += e[o]; }
        for (int o = 0; o < OUT_DIM; ++o) out[b * OUT_DIM + o] = e[o] / sum;
    }
}

// ---------------------------------------------------------------------------
// Host-side orchestration
// ---------------------------------------------------------------------------
static inline int gemm_grid(int M, int N, int msub, int nsub) {
    const int waves = (M / (16 * msub)) * (N / (16 * nsub));
    return (waves + 7) / 8;                 // 8 waves per 256-thread block
}

extern "C" void kernel_launch(void* const* d_in, const int* in_sizes, int n_in,
                              void* d_out, int out_size, void* d_ws, size_t ws_size,
                              hipStream_t stream)
{
    (void)in_sizes; (void)n_in; (void)out_size; (void)ws_size;

    const float* x         = (const float*)d_in[0];   // [B,L,128]
    const float* enc_w     = (const float*)d_in[1];   // [256,128]
    const float* enc_b     = (const float*)d_in[2];   // [256]
    const float* ln_w      = (const float*)d_in[3];   // [nb,256]
    const float* ln_b      = (const float*)d_in[4];   // [nb,256]
    const float* in_proj_w = (const float*)d_in[5];   // [nb,1024,256]
    const float* conv_w    = (const float*)d_in[6];   // [nb,512,4]
    const float* conv_b    = (const float*)d_in[7];   // [nb,512]
    const float* xproj_w   = (const float*)d_in[8];   // [nb,48,512]
    const float* dtproj_w  = (const float*)d_in[9];   // [nb,512,16]
    const float* dtproj_b  = (const float*)d_in[10];  // [nb,512]
    const float* A_log     = (const float*)d_in[11];  // [nb,512,16]
    const float* Dp        = (const float*)d_in[12];  // [nb,512]
    const float* outproj_w = (const float*)d_in[13];  // [nb,256,512]
    const float* dec_w     = (const float*)d_in[14];  // [10,256]
    const float* dec_b     = (const float*)d_in[15];  // [10]
    float* out = (float*)d_out;

    // Workspace layout (fp32)
    float* ws = (float*)d_ws;
    size_t off = 0;
    float* h   = ws + off; off += (size_t)BL * D_MODEL;        // residual stream
    float* nrm = ws + off; off += (size_t)BL * D_MODEL;        // layernorm output
    float* xz  = ws + off; off += (size_t)BL * 2 * D_INNER;    // in_proj output
    float* xi  = ws + off; off += (size_t)BL * D_INNER;        // conv+silu output
    float* dbl = ws + off; off += (size_t)BL * (DT_RANK + 2 * D_STATE);
    float* dy  = ws + off; off += (size_t)BL * D_INNER;        // delta, then y*silu(z)

    const dim3 blk(256);

    // Encoder: h = x @ enc_w^T + enc_b
    gemm_wmma_f32<EPI_BIAS, 2, 4><<<gemm_grid(BL, D_MODEL, 2, 4), blk, 0, stream>>>(
        x, IN_DIM, enc_w, IN_DIM, enc_b, nullptr, 0, h, D_MODEL, BL, D_MODEL, IN_DIM);

    for (int i = 0; i < NUM_BLOCKS; ++i) {
        // LayerNorm
        ln_kernel<<<(BL * 32 + 255) / 256, blk, 0, stream>>>(
            h, ln_w + i * D_MODEL, ln_b + i * D_MODEL, nrm, BL);

        // in_proj: xz = nrm @ W^T   [BL, 1024]
        gemm_wmma_f32<EPI_NONE, 2, 4><<<gemm_grid(BL, 2 * D_INNER, 2, 4), blk, 0, stream>>>(
            nrm, D_MODEL, in_proj_w + (size_t)i * 2 * D_INNER * D_MODEL, D_MODEL,
            nullptr, nullptr, 0, xz, 2 * D_INNER, BL, 2 * D_INNER, D_MODEL);

        // depthwise causal conv + SiLU -> xi
        conv_silu_kernel<<<((size_t)BL * D_INNER + 255) / 256, blk, 0, stream>>>(
            xz, conv_w + (size_t)i * D_INNER * D_CONV, conv_b + i * D_INNER, xi);

        // xproj: dbl = xi @ W^T   [BL, 48]
        gemm_wmma_f32<EPI_NONE, 2, 3><<<gemm_grid(BL, DT_RANK + 2 * D_STATE, 2, 3), blk, 0, stream>>>(
            xi, D_INNER, xproj_w + (size_t)i * (DT_RANK + 2 * D_STATE) * D_INNER, D_INNER,
            nullptr, nullptr, 0, dbl, DT_RANK + 2 * D_STATE,
            BL, DT_RANK + 2 * D_STATE, D_INNER);

        // dtproj: dy = softplus(dt @ W^T + b)   [BL, 512]; dt = dbl[:, :16] (lda=48)
        gemm_wmma_f32<EPI_SOFTPLUS, 2, 4><<<gemm_grid(BL, D_INNER, 2, 4), blk, 0, stream>>>(
            dbl, DT_RANK + 2 * D_STATE, dtproj_w + (size_t)i * D_INNER * DT_RANK, DT_RANK,
            dtproj_b + i * D_INNER, nullptr, 0, dy, D_INNER, BL, D_INNER, DT_RANK);

        // selective scan (in place: dy: delta -> y*silu(z))
        scan_kernel<<<((BATCH * D_INNER / 2) * 32 + 255) / 256, blk, 0, stream>>>(
            dy, xi, dbl, xz, A_log + (size_t)i * D_INNER * D_STATE, Dp + i * D_INNER);

        // outproj + fused gelu + residual, in place into h
        gemm_wmma_f32<EPI_GELU_SKIP, 2, 4><<<gemm_grid(BL, D_MODEL, 2, 4), blk, 0, stream>>>(
            dy, D_INNER, outproj_w + (size_t)i * D_MODEL * D_INNER, D_INNER,
            nullptr, h, D_MODEL, h, D_MODEL, BL, D_MODEL, D_INNER);
    }

    // mean pool + decoder + softmax
    head_kernel<<<BATCH, D_MODEL, 0, stream>>>(h, dec_w, dec_b, out);
}